// GNNCritic_48653389529336
// MI455X (gfx1250) — compile-verified
//
#include <hip/hip_runtime.h>
#include <math.h>

typedef __attribute__((ext_vector_type(2))) float v2f;
typedef __attribute__((ext_vector_type(8))) float v8f;

#define F_NODE 64
#define F_ACT  16
#define HDIM   128
#define NEG_SLOPE 0.2f

__device__ __forceinline__ float wave_sum(float v) {
#pragma unroll
  for (int m = 16; m >= 1; m >>= 1) v += __shfl_xor(v, m, 32);
  return v;
}
__device__ __forceinline__ float wave_max(float v) {
#pragma unroll
  for (int m = 16; m >= 1; m >>= 1) v = fmaxf(v, __shfl_xor(v, m, 32));
  return v;
}
__device__ __forceinline__ float lrelu(float x) { return x >= 0.f ? x : NEG_SLOPE * x; }

// ---------------- CSR construction ----------------
__global__ void init_kernel(unsigned* deg, unsigned* fill, float* a0, float* a1, int N) {
  int i = blockIdx.x * 256 + threadIdx.x;
  if (i < N) { deg[i] = 0u; fill[i] = 0u; a0[i] = 0.f; a1[i] = 0.f; }
}

__global__ void count_kernel(const int* __restrict__ ei, const float* __restrict__ eattr,
                             unsigned* deg, float* a0, float* a1, int E) {
  int e = blockIdx.x * 256 + threadIdx.x;
  if (e >= E) return;
  int dst = ei[E + e];
  atomicAdd(&deg[dst], 1u);
  atomicAdd(&a0[dst], eattr[2 * e + 0]);
  atomicAdd(&a1[dst], eattr[2 * e + 1]);
}

__global__ void mean_kernel(const unsigned* deg, float* a0, float* a1, int N) {
  int i = blockIdx.x * 256 + threadIdx.x;
  if (i >= N) return;
  float d = (float)deg[i];
  float inv = 1.0f / fmaxf(d, 1.0f);
  a0[i] *= inv; a1[i] *= inv;
}

__global__ void scan_kernel(const unsigned* deg, unsigned* rptr, int n) {
  __shared__ unsigned temp[1024];
  __shared__ unsigned running;
  if (threadIdx.x == 0) running = 0u;
  __syncthreads();
  for (int base = 0; base < n; base += 1024) {
    int i = base + (int)threadIdx.x;
    unsigned v = (i < n) ? (deg[i] + 1u) : 0u;  // +1 self-loop
    temp[threadIdx.x] = v;
    __syncthreads();
    for (int o = 1; o < 1024; o <<= 1) {
      unsigned t = (threadIdx.x >= (unsigned)o) ? temp[threadIdx.x - o] : 0u;
      __syncthreads();
      temp[threadIdx.x] += t;
      __syncthreads();
    }
    if (i < n) rptr[i] = running + temp[threadIdx.x] - v;  // exclusive
    __syncthreads();
    if (threadIdx.x == 0) running += temp[1023];
    __syncthreads();
  }
  if (threadIdx.x == 0) rptr[n] = running;
}

__global__ void fill_kernel(const int* __restrict__ ei, const float* __restrict__ eattr,
                            const unsigned* rptr, unsigned* fill,
                            int* csrc, float* cea0, float* cea1, int E) {
  int e = blockIdx.x * 256 + threadIdx.x;
  if (e >= E) return;
  int src = ei[e], dst = ei[E + e];
  unsigned pos = rptr[dst] + atomicAdd(&fill[dst], 1u);
  csrc[pos] = src;
  cea0[pos] = eattr[2 * e + 0];
  cea1[pos] = eattr[2 * e + 1];
}

__global__ void selfloop_kernel(const unsigned* rptr, const float* a0, const float* a1,
                                int* csrc, float* cea0, float* cea1, int N) {
  int i = blockIdx.x * 256 + threadIdx.x;
  if (i >= N) return;
  unsigned pos = rptr[i + 1] - 1u;  // last slot of row = self-loop
  csrc[pos] = i;
  cea0[pos] = a0[i];
  cea1[pos] = a1[i];
}

// ---------------- per-layer kernels ----------------
__global__ void we_kernel(const float* __restrict__ We, const float* __restrict__ ae, float* wev) {
  int lane = threadIdx.x & 31;
  float s0 = 0.f, s1 = 0.f;
  for (int c = lane; c < HDIM; c += 32) {
    float a = ae[c];
    s0 += We[c] * a;
    s1 += We[HDIM + c] * a;
  }
  s0 = wave_sum(s0); s1 = wave_sum(s1);
  if (lane == 0) { wev[0] = s0; wev[1] = s1; }
}

// MODE 0: A = [x | action] (K=80). MODE 1: A = prev (K=128). W: [K,128] row-major.
// LDS holds W pair-interleaved: sW2[q*256 + 2n + p] = W[(2q+p)*128 + n]
// so each lane's B fragment (k = k0+2*half+{0,1}, n fixed) is one aligned b64 load.
template <int MODE>
__global__ __launch_bounds__(256) void gemm_wmma_kernel(
    const float* __restrict__ A, const float* __restrict__ action,
    const float* __restrict__ W, float* __restrict__ H, int N, int K) {
  __shared__ float sW2[HDIM * HDIM];  // up to 128x128 = 64KB
  for (int idx = threadIdx.x; idx < K * HDIM; idx += 256) {
    int k = idx >> 7;        // idx / 128
    int n = idx & 127;       // idx % 128
    sW2[(k >> 1) * 256 + 2 * n + (k & 1)] = W[idx];
  }
  __syncthreads();

  const int wave = threadIdx.x >> 5;
  const int lane = threadIdx.x & 31;
  const int laneM = lane & 15;
  const int half = lane >> 4;
  const int row0 = blockIdx.x * 128 + wave * 16;
  const int arow = row0 + laneM;
  const bool rowOK = arow < N;

  const v2f zero2 = {0.f, 0.f};
  v8f zero = {0.f, 0.f, 0.f, 0.f, 0.f, 0.f, 0.f, 0.f};
  v8f acc[8];
#pragma unroll
  for (int t = 0; t < 8; t++) acc[t] = zero;

  for (int k0 = 0; k0 < K; k0 += 4) {
    v2f a;
    if (MODE == 0) {
      if (k0 < F_NODE) {
        a = rowOK ? *(const v2f*)(A + (size_t)arow * F_NODE + k0 + 2 * half) : zero2;
      } else {
        // action columns are row-independent; rows >= N never get stored
        a = *(const v2f*)(action + (k0 - F_NODE) + 2 * half);
      }
    } else {
      a = rowOK ? *(const v2f*)(A + (size_t)arow * HDIM + k0 + 2 * half) : zero2;
    }
    const float* bbase = &sW2[((k0 >> 1) + half) * 256 + 2 * laneM];
#pragma unroll
    for (int t = 0; t < 8; t++) {
      v2f b = *(const v2f*)(bbase + t * 32);
      acc[t] = __builtin_amdgcn_wmma_f32_16x16x4_f32(false, a, false, b,
                                                     (short)0, acc[t], false, false);
    }
  }
#pragma unroll
  for (int t = 0; t < 8; t++) {
#pragma unroll
    for (int v = 0; v < 8; ++v) {
      int r = row0 + v + 8 * half;
      if (r < N) H[(size_t)r * HDIM + t * 16 + laneM] = acc[t][v];
    }
  }
}

__global__ __launch_bounds__(256) void rowdot_kernel(const float* __restrict__ h,
                                                     const float* __restrict__ asrc,
                                                     const float* __restrict__ adst,
                                                     float* als, float* ald, int N) {
  int node = blockIdx.x * 8 + (threadIdx.x >> 5);
  if (node >= N) return;
  int lane = threadIdx.x & 31;
  const float4* h4 = (const float4*)h;
  float4 hv = h4[(size_t)node * 32 + lane];
  float4 sv = ((const float4*)asrc)[lane];
  float4 dv = ((const float4*)adst)[lane];
  float ss = hv.x * sv.x + hv.y * sv.y + hv.z * sv.z + hv.w * sv.w;
  float sd = hv.x * dv.x + hv.y * dv.y + hv.z * dv.z + hv.w * dv.w;
  ss = wave_sum(ss); sd = wave_sum(sd);
  if (lane == 0) { als[node] = ss; ald[node] = sd; }
}

__global__ __launch_bounds__(256) void agg_kernel(
    const unsigned* __restrict__ rptr, const int* __restrict__ csrc,
    const float* __restrict__ cea0, const float* __restrict__ cea1,
    const float* __restrict__ als, const float* __restrict__ ald,
    const float* __restrict__ wev, const float* __restrict__ h,
    const float* __restrict__ bias, float* __restrict__ out, int N) {
  int node = blockIdx.x * 8 + (threadIdx.x >> 5);
  if (node >= N) return;
  int lane = threadIdx.x & 31;
  unsigned base = rptr[node];
  int cnt = (int)(rptr[node + 1] - base);
  float we0 = wev[0], we1 = wev[1];
  float aldi = ald[node];

  // pass 1: row max of leaky-relu scores
  float mx = -3.0e38f;
  for (int j = lane; j < cnt; j += 32) {
    int s = csrc[base + j];
    float sc = lrelu(als[s] + aldi + cea0[base + j] * we0 + cea1[base + j] * we1);
    mx = fmaxf(mx, sc);
  }
  mx = wave_max(mx);

  // pass 2: softmax denominator
  float den = 0.f;
  for (int j = lane; j < cnt; j += 32) {
    int s = csrc[base + j];
    float sc = lrelu(als[s] + aldi + cea0[base + j] * we0 + cea1[base + j] * we1);
    den += expf(sc - mx);
  }
  den = wave_sum(den);
  float inv = 1.0f / (den + 1e-16f);

  // pass 3: weighted feature gather (lane owns 4 of 128 columns)
  const float4* h4 = (const float4*)h;
  float4 acc = {0.f, 0.f, 0.f, 0.f};
  for (int j = 0; j < cnt; j++) {
    int s = csrc[base + j];
    float sc = lrelu(als[s] + aldi + cea0[base + j] * we0 + cea1[base + j] * we1);
    float coef = expf(sc - mx) * inv;
    float4 hv = h4[(size_t)s * 32 + lane];
    acc.x += coef * hv.x; acc.y += coef * hv.y;
    acc.z += coef * hv.z; acc.w += coef * hv.w;
  }
  float4 bb = ((const float4*)bias)[lane];
  float4 o;
  o.x = fmaxf(acc.x + bb.x, 0.f);
  o.y = fmaxf(acc.y + bb.y, 0.f);
  o.z = fmaxf(acc.z + bb.z, 0.f);
  o.w = fmaxf(acc.w + bb.w, 0.f);
  ((float4*)out)[(size_t)node * 32 + lane] = o;
}

// ---------------- pooling + fc head ----------------
__global__ void pool_partial_kernel(const float* __restrict__ x, float* partial, int N) {
  int c = threadIdx.x;  // 128
  float s = 0.f;
  for (int i = blockIdx.x; i < N; i += gridDim.x) s += x[(size_t)i * HDIM + c];
  partial[blockIdx.x * HDIM + c] = s;
}

__global__ void pool_final_kernel(const float* __restrict__ partial,
                                  const float* __restrict__ fcW,
                                  const float* __restrict__ fcb,
                                  float* out, int N, int P) {
  int c = threadIdx.x;  // 128
  float s = 0.f;
  for (int b = 0; b < P; b++) s += partial[b * HDIM + c];
  s = (s / (float)N) * fcW[c];
  __shared__ float red[HDIM];
  red[c] = s;
  __syncthreads();
  for (int o = 64; o >= 1; o >>= 1) {
    if (c < o) red[c] += red[c + o];
    __syncthreads();
  }
  if (c == 0) out[0] = red[0] + fcb[0];
}

extern "C" void kernel_launch(void* const* d_in, const int* in_sizes, int n_in,
                              void* d_out, int out_size, void* d_ws, size_t ws_size,
                              hipStream_t stream) {
  (void)n_in; (void)out_size; (void)ws_size;
  const float* x      = (const float*)d_in[0];
  const float* eattr  = (const float*)d_in[1];
  const float* action = (const float*)d_in[2];
  const int*   ei     = (const int*)d_in[3];
  const float* W1     = (const float*)d_in[4];
  const float* asrc1  = (const float*)d_in[5];
  const float* adst1  = (const float*)d_in[6];
  const float* We1    = (const float*)d_in[7];
  const float* ae1    = (const float*)d_in[8];
  const float* b1     = (const float*)d_in[9];
  const float* W2     = (const float*)d_in[10];
  const float* asrc2  = (const float*)d_in[11];
  const float* adst2  = (const float*)d_in[12];
  const float* We2    = (const float*)d_in[13];
  const float* ae2    = (const float*)d_in[14];
  const float* b2     = (const float*)d_in[15];
  const float* fcW    = (const float*)d_in[16];
  const float* fcb    = (const float*)d_in[17];

  const int N  = in_sizes[0] / F_NODE;
  const int E  = in_sizes[3] / 2;
  const int EP = E + N;

  char* ws = (char*)d_ws;
  size_t off = 0;
  auto alloc = [&](size_t bytes) -> char* {
    off = (off + 255) & ~(size_t)255;
    char* p = ws + off;
    off += bytes;
    return p;
  };
  unsigned* deg  = (unsigned*)alloc((size_t)N * 4);
  unsigned* fill = (unsigned*)alloc((size_t)N * 4);
  unsigned* rptr = (unsigned*)alloc((size_t)(N + 1) * 4);
  float* a0      = (float*)alloc((size_t)N * 4);
  float* a1      = (float*)alloc((size_t)N * 4);
  int*   csrc    = (int*)alloc((size_t)EP * 4);
  float* cea0    = (float*)alloc((size_t)EP * 4);
  float* cea1    = (float*)alloc((size_t)EP * 4);
  float* hbuf    = (float*)alloc((size_t)N * HDIM * 4);
  float* xbuf    = (float*)alloc((size_t)N * HDIM * 4);
  float* als     = (float*)alloc((size_t)N * 4);
  float* ald     = (float*)alloc((size_t)N * 4);
  float* wev     = (float*)alloc(8);
  float* partial = (float*)alloc((size_t)256 * HDIM * 4);

  int gN = (N + 255) / 256, gE = (E + 255) / 256;
  init_kernel<<<gN, 256, 0, stream>>>(deg, fill, a0, a1, N);
  count_kernel<<<gE, 256, 0, stream>>>(ei, eattr, deg, a0, a1, E);
  mean_kernel<<<gN, 256, 0, stream>>>(deg, a0, a1, N);
  scan_kernel<<<1, 1024, 0, stream>>>(deg, rptr, N);
  fill_kernel<<<gE, 256, 0, stream>>>(ei, eattr, rptr, fill, csrc, cea0, cea1, E);
  selfloop_kernel<<<gN, 256, 0, stream>>>(rptr, a0, a1, csrc, cea0, cea1, N);

  // layer 1
  we_kernel<<<1, 32, 0, stream>>>(We1, ae1, wev);
  gemm_wmma_kernel<0><<<(N + 127) / 128, 256, 0, stream>>>(x, action, W1, hbuf, N, F_NODE + F_ACT);
  rowdot_kernel<<<(N + 7) / 8, 256, 0, stream>>>(hbuf, asrc1, adst1, als, ald, N);
  agg_kernel<<<(N + 7) / 8, 256, 0, stream>>>(rptr, csrc, cea0, cea1, als, ald, wev, hbuf, b1, xbuf, N);

  // layer 2
  we_kernel<<<1, 32, 0, stream>>>(We2, ae2, wev);
  gemm_wmma_kernel<1><<<(N + 127) / 128, 256, 0, stream>>>(xbuf, nullptr, W2, hbuf, N, HDIM);
  rowdot_kernel<<<(N + 7) / 8, 256, 0, stream>>>(hbuf, asrc2, adst2, als, ald, N);
  agg_kernel<<<(N + 7) / 8, 256, 0, stream>>>(rptr, csrc, cea0, cea1, als, ald, wev, hbuf, b2, xbuf, N);

  // mean pool + fc
  pool_partial_kernel<<<256, HDIM, 0, stream>>>(xbuf, partial, N);
  pool_final_kernel<<<1, HDIM, 0, stream>>>(partial, fcW, fcb, (float*)d_out, N, 256);
}